// MultiBoxLoss_75359496175904
// MI455X (gfx1250) — compile-verified
//
#include <hip/hip_runtime.h>
#include <hip/hip_bf16.h>
#include <stdint.h>

// MultiBoxLoss (RetinaFace-style) for MI455X / gfx1250.
// Memory-bound elementwise+reduction+select workload: no GEMM content, so the
// CDNA5 win is data movement (async global->LDS staging of GT tiles, wave32
// sizing, split wait counters, prefetch), not WMMA.

#define Bn 32
#define Pn 65536
#define Mn 32
#define THRESHF 0.35f

__device__ __forceinline__ float sl1(float d) {
    float a = fabsf(d);
    return (a < 1.0f) ? 0.5f * d * d : a - 0.5f;
}

// One b128 per lane of wave 0, global -> LDS via the gfx1250 async engine.
// Caller must s_wait_asynccnt + __syncthreads before other waves read the LDS.
__device__ __forceinline__ void async_copy_b128_wave0(unsigned lds_byte_addr,
                                                      unsigned long long gaddr) {
    asm volatile("global_load_async_to_lds_b128 %0, %1, off"
                 :: "v"(lds_byte_addr), "v"(gaddr) : "memory");
}

__device__ __forceinline__ void wait_async0() {
    asm volatile("s_wait_asynccnt 0x0" ::: "memory");
}

// ---------------------------------------------------------------- matching ---
__global__ __launch_bounds__(256) void k_match(const float* __restrict__ priors,
                                               const float* __restrict__ boxes_gt,
                                               unsigned char* __restrict__ match,
                                               unsigned long long* __restrict__ keys) {
    __shared__ float tb[Mn * 4];                 // 512B GT boxes (point form)
    __shared__ unsigned long long smax[Mn];      // per-truth best (key) in block
    const int b = blockIdx.y;
    const int tid = threadIdx.x;

    if (tid < 32) {
        unsigned lds = (unsigned)(size_t)(&tb[0]) + (unsigned)tid * 16u;
        unsigned long long ga =
            (unsigned long long)(size_t)(boxes_gt + (size_t)b * Mn * 4) + (unsigned long long)tid * 16ull;
        async_copy_b128_wave0(lds, ga);
        smax[tid] = 0ull;
        wait_async0();
    }
    __syncthreads();

    const int p = blockIdx.x * 256 + tid;
    const float4 pr = ((const float4*)priors)[p];       // center form cx,cy,w,h
    const float px1 = pr.x - 0.5f * pr.z, py1 = pr.y - 0.5f * pr.w;
    const float px2 = pr.x + 0.5f * pr.z, py2 = pr.y + 0.5f * pr.w;
    const float area_p = pr.z * pr.w;

    float best = -1.0f;
    int bidx = 0;
    const unsigned invp = 0xFFFFFFFFu - (unsigned)p;    // smaller p wins ties

#pragma unroll 4
    for (int m = 0; m < Mn; ++m) {
        const float tx1 = tb[m * 4 + 0], ty1 = tb[m * 4 + 1];
        const float tx2 = tb[m * 4 + 2], ty2 = tb[m * 4 + 3];
        float iw = fminf(tx2, px2) - fmaxf(tx1, px1);
        float ih = fminf(ty2, py2) - fmaxf(ty1, py1);
        iw = fmaxf(iw, 0.0f);
        ih = fmaxf(ih, 0.0f);
        const float inter = iw * ih;
        const float at = (tx2 - tx1) * (ty2 - ty1);
        const float iou = inter / (at + area_p - inter);
        if (iou > best) { best = iou; bidx = m; }       // first-max (argmax axis=0)
        const unsigned long long key =
            ((unsigned long long)__float_as_uint(iou) << 32) | (unsigned long long)invp;
        atomicMax(&smax[m], key);
    }

    match[(size_t)b * Pn + p] =
        (unsigned char)(bidx | ((best >= THRESHF) ? 0x80 : 0x00));

    __syncthreads();
    if (tid < Mn) atomicMax(&keys[b * Mn + tid], smax[tid]);
}

// best-prior override: serial over m per batch => last-m-wins on collisions
__global__ void k_override(const unsigned long long* __restrict__ keys,
                           unsigned char* __restrict__ match) {
    const int b = threadIdx.x;
    if (b >= Bn) return;
    for (int m = 0; m < Mn; ++m) {
        const unsigned long long key = keys[b * Mn + m];
        const unsigned p = 0xFFFFFFFFu - (unsigned)(key & 0xFFFFFFFFull);
        match[(size_t)b * Pn + p] = (unsigned char)(0x80 | m);  // forced positive
    }
}

// ------------------------- losses + fused level-1 select histogram -----------
__global__ __launch_bounds__(256) void k_losses(const float* __restrict__ loc,
                                                const float* __restrict__ conf,
                                                const float* __restrict__ landm,
                                                const float* __restrict__ priors,
                                                const float* __restrict__ boxes_gt,
                                                const float* __restrict__ kp_gt,
                                                const unsigned char* __restrict__ match,
                                                float* __restrict__ lossc,
                                                unsigned* __restrict__ h1,
                                                float* __restrict__ p_ll,
                                                float* __restrict__ p_lm,
                                                float* __restrict__ p_ce,
                                                int* __restrict__ numpos) {
    __shared__ float tb[Mn * 4];    // 512B boxes
    __shared__ float tk[Mn * 10];   // 1280B keypoints
    __shared__ unsigned lh[4096];   // level-1 histogram (bits >> 20)
    __shared__ float rf[256];
    __shared__ int ri[256];
    const int b = blockIdx.y, chunk = blockIdx.x, tid = threadIdx.x;

    if (tid < 32) {
        const unsigned ldsb = (unsigned)(size_t)(&tb[0]) + (unsigned)tid * 16u;
        const unsigned ldsk = (unsigned)(size_t)(&tk[0]) + (unsigned)tid * 16u;
        const unsigned long long gb =
            (unsigned long long)(size_t)(boxes_gt + (size_t)b * Mn * 4) + (unsigned long long)tid * 16ull;
        const unsigned long long gk =
            (unsigned long long)(size_t)(kp_gt + (size_t)b * Mn * 10) + (unsigned long long)tid * 16ull;
        async_copy_b128_wave0(ldsb, gb);                 // 512B
        async_copy_b128_wave0(ldsk, gk);                 // kp bytes 0..511
        async_copy_b128_wave0(ldsk + 512u, gk + 512ull); // kp bytes 512..1023
        if (tid < 16)
            async_copy_b128_wave0(ldsk + 1024u, gk + 1024ull); // kp 1024..1279
    }
    for (int i = tid; i < 4096; i += 256) lh[i] = 0;
    if (tid < 32) wait_async0();
    __syncthreads();

    float ll = 0.f, lm = 0.f, pce = 0.f;
    int np = 0;
    const size_t rowbase = (size_t)b * Pn;
    const float2* conf2 = (const float2*)conf;

    for (int k = 0; k < 8; ++k) {
        const int p = chunk * 2048 + k * 256 + tid;
        // prefetch the always-read conf stream a few iterations ahead
        // (speculative prefetch: OOB at the tail is silently dropped)
        __builtin_prefetch(conf2 + rowbase + p + 1024, 0, 0);

        const unsigned char code = match[rowbase + p];
        const bool pos = (code & 0x80) != 0;
        const int idx = code & 31;

        const float2 cf = conf2[rowbase + p];
        const float mx = fmaxf(cf.x, cf.y);
        const float mn = fminf(cf.x, cf.y);
        const float lse = mx + log1pf(expf(mn - mx));
        const float ce = lse - (pos ? cf.y : cf.x);      // labels are all 1

        const float lc_v = pos ? 0.0f : ce;
        lossc[rowbase + p] = lc_v;
        atomicAdd(&lh[__float_as_uint(lc_v) >> 20], 1u); // ce >= 0 -> bit-monotone

        if (pos) {
            pce += ce;
            np += 1;
            const float4 pr = ((const float4*)priors)[p];
            const float iw = 10.0f / pr.z;               // 1/(VAR0*w)
            const float ih = 10.0f / pr.w;
            // encode matched box
            const float bx1 = tb[idx * 4 + 0], by1 = tb[idx * 4 + 1];
            const float bx2 = tb[idx * 4 + 2], by2 = tb[idx * 4 + 3];
            const float gx = ((bx1 + bx2) * 0.5f - pr.x) * iw;
            const float gy = ((by1 + by2) * 0.5f - pr.y) * ih;
            const float gw = logf((bx2 - bx1) / pr.z) * 5.0f;  // /VAR1
            const float gh = logf((by2 - by1) / pr.w) * 5.0f;
            const float4 lc = ((const float4*)loc)[rowbase + p];
            ll += sl1(lc.x - gx) + sl1(lc.y - gy) + sl1(lc.z - gw) + sl1(lc.w - gh);
            // encode matched landmarks (5 points)
            const float2* ld2 = (const float2*)landm + (rowbase + p) * 5;
#pragma unroll
            for (int j = 0; j < 5; ++j) {
                const float2 l = ld2[j];
                const float ex = (tk[idx * 10 + 2 * j + 0] - pr.x) * iw;
                const float ey = (tk[idx * 10 + 2 * j + 1] - pr.y) * ih;
                lm += sl1(ex - l.x) + sl1(ey - l.y);
            }
        }
    }

    __syncthreads();
    // merge block histogram into the per-batch global histogram
    for (int i = tid; i < 4096; i += 256) {
        const unsigned c = lh[i];
        if (c) atomicAdd(&h1[b * 4096 + i], c);
    }

    // deterministic block tree reductions
    rf[tid] = ll; __syncthreads();
    for (int s = 128; s > 0; s >>= 1) { if (tid < s) rf[tid] += rf[tid + s]; __syncthreads(); }
    if (tid == 0) p_ll[b * 32 + chunk] = rf[0];
    __syncthreads();
    rf[tid] = lm; __syncthreads();
    for (int s = 128; s > 0; s >>= 1) { if (tid < s) rf[tid] += rf[tid + s]; __syncthreads(); }
    if (tid == 0) p_lm[b * 32 + chunk] = rf[0];
    __syncthreads();
    rf[tid] = pce; __syncthreads();
    for (int s = 128; s > 0; s >>= 1) { if (tid < s) rf[tid] += rf[tid + s]; __syncthreads(); }
    if (tid == 0) p_ce[b * 32 + chunk] = rf[0];
    __syncthreads();
    ri[tid] = np; __syncthreads();
    for (int s = 128; s > 0; s >>= 1) { if (tid < s) ri[tid] += ri[tid + s]; __syncthreads(); }
    if (tid == 0) atomicAdd(&numpos[b], ri[0]);          // int atomic: deterministic
}

// -------------------------------------------------- exact top-K radix select --
__global__ void k_scan1(const unsigned* __restrict__ h1, const int* __restrict__ numpos,
                        int* __restrict__ t1a, int* __restrict__ k1a,
                        unsigned* __restrict__ Tfin, int* __restrict__ eq) {
    const int b = threadIdx.x;
    if (b >= Bn) return;
    long long K = 7ll * (long long)numpos[b];
    if (K > Pn - 1) K = Pn - 1;                          // clip a_max=P-1
    if (K == 0) { t1a[b] = -1; k1a[b] = 0; Tfin[b] = 0xFFFFFFFFu; eq[b] = 0; return; }
    long long cum = 0;
    for (int bin = 4095; bin >= 0; --bin) {
        const unsigned c = h1[b * 4096 + bin];
        if (cum + (long long)c >= K) { t1a[b] = bin; k1a[b] = (int)(K - cum); return; }
        cum += c;
    }
    t1a[b] = 0; k1a[b] = (int)(K - cum);
}

__global__ __launch_bounds__(256) void k_hist2(const float* __restrict__ lossc,
                                               const int* __restrict__ t1a,
                                               unsigned* __restrict__ h2) {
    const int b = blockIdx.y, chunk = blockIdx.x, tid = threadIdx.x;
    const int t1 = t1a[b];
    if (t1 < 0) return;
    const size_t rowbase = (size_t)b * Pn;
    for (int k = 0; k < 8; ++k) {
        const int p = chunk * 2048 + k * 256 + tid;
        const unsigned bits = __float_as_uint(lossc[rowbase + p]);
        if ((int)(bits >> 20) == t1)
            atomicAdd(&h2[b * 4096 + ((bits >> 8) & 0xFFFu)], 1u);
    }
}

__global__ void k_scan2(const unsigned* __restrict__ h2, const int* __restrict__ t1a,
                        const int* __restrict__ k1a, int* __restrict__ t12a,
                        int* __restrict__ k2a) {
    const int b = threadIdx.x;
    if (b >= Bn) return;
    const int t1 = t1a[b];
    if (t1 < 0) { t12a[b] = -1; k2a[b] = 0; return; }
    const long long K = k1a[b];
    long long cum = 0;
    for (int bin = 4095; bin >= 0; --bin) {
        const unsigned c = h2[b * 4096 + bin];
        if (cum + (long long)c >= K) { t12a[b] = (t1 << 12) | bin; k2a[b] = (int)(K - cum); return; }
        cum += c;
    }
    t12a[b] = t1 << 12; k2a[b] = (int)(K - cum);
}

__global__ __launch_bounds__(256) void k_hist3(const float* __restrict__ lossc,
                                               const int* __restrict__ t12a,
                                               unsigned* __restrict__ h3) {
    const int b = blockIdx.y, chunk = blockIdx.x, tid = threadIdx.x;
    const int t12 = t12a[b];
    if (t12 < 0) return;
    const size_t rowbase = (size_t)b * Pn;
    for (int k = 0; k < 8; ++k) {
        const int p = chunk * 2048 + k * 256 + tid;
        const unsigned bits = __float_as_uint(lossc[rowbase + p]);
        if ((int)(bits >> 8) == t12)
            atomicAdd(&h3[b * 256 + (bits & 0xFFu)], 1u);
    }
}

__global__ void k_scan3(const unsigned* __restrict__ h3, const int* __restrict__ t12a,
                        const int* __restrict__ k2a, unsigned* __restrict__ Tfin,
                        int* __restrict__ eq) {
    const int b = threadIdx.x;
    if (b >= Bn) return;
    const int t12 = t12a[b];
    if (t12 < 0) return;                                 // K==0 already handled
    const long long K = k2a[b];
    long long cum = 0;
    for (int bin = 255; bin >= 0; --bin) {
        const unsigned c = h3[b * 256 + bin];
        if (cum + (long long)c >= K) {
            Tfin[b] = ((unsigned)t12 << 8) | (unsigned)bin;
            eq[b] = (int)(K - cum);
            return;
        }
        cum += c;
    }
    Tfin[b] = (unsigned)t12 << 8; eq[b] = 0;
}

__global__ __launch_bounds__(256) void k_negsum(const float* __restrict__ lossc,
                                                const unsigned* __restrict__ Tfin,
                                                float* __restrict__ p_neg) {
    __shared__ float rf[256];
    const int b = blockIdx.y, chunk = blockIdx.x, tid = threadIdx.x;
    const unsigned T = Tfin[b];
    const size_t rowbase = (size_t)b * Pn;
    float acc = 0.f;
    for (int k = 0; k < 8; ++k) {
        const int p = chunk * 2048 + k * 256 + tid;
        const float v = lossc[rowbase + p];
        if (__float_as_uint(v) > T) acc += v;            // strictly above K-th value
    }
    rf[tid] = acc; __syncthreads();
    for (int s = 128; s > 0; s >>= 1) { if (tid < s) rf[tid] += rf[tid + s]; __syncthreads(); }
    if (tid == 0) p_neg[b * 32 + chunk] = rf[0];
}

__global__ void k_combine(const float* __restrict__ p_ll, const float* __restrict__ p_lm,
                          const float* __restrict__ p_ce, const float* __restrict__ p_neg,
                          const int* __restrict__ numpos, const unsigned* __restrict__ Tfin,
                          const int* __restrict__ eq, float* __restrict__ out) {
    if (threadIdx.x != 0 || blockIdx.x != 0) return;     // single-thread: deterministic
    float sll = 0.f, slm = 0.f, sce = 0.f, sneg = 0.f;
    for (int i = 0; i < Bn * 32; ++i) {
        sll += p_ll[i]; slm += p_lm[i]; sce += p_ce[i]; sneg += p_neg[i];
    }
    long long tp = 0;
    for (int b = 0; b < Bn; ++b) {
        tp += numpos[b];
        const int e = eq[b];
        if (e > 0) sneg += (float)e * __uint_as_float(Tfin[b]); // ties at K-th value
    }
    const float n = fmaxf((float)tp, 1.0f);
    const float n1 = fmaxf((float)tp * 10.0f, 1.0f);
    const float loss_l = sll / n;
    const float loss_cls = (sce + sneg) / n;
    const float loss_landm = slm / n1;
    out[0] = 2.0f * loss_l + loss_cls + loss_landm;      // W_LOC=2, W_CLS=W_LND=1
    out[1] = loss_l;
    out[2] = loss_cls;
    out[3] = loss_landm;
}

// ------------------------------------------------------------------- launch ---
extern "C" void kernel_launch(void* const* d_in, const int* in_sizes, int n_in,
                              void* d_out, int out_size, void* d_ws, size_t ws_size,
                              hipStream_t stream) {
    (void)in_sizes; (void)n_in; (void)out_size; (void)ws_size;
    const float* loc      = (const float*)d_in[0];   // [B,P,4]
    const float* conf     = (const float*)d_in[1];   // [B,P,2]
    const float* landm    = (const float*)d_in[2];   // [B,P,10]
    const float* priors   = (const float*)d_in[3];   // [P,4] center form
    const float* boxes_gt = (const float*)d_in[4];   // [B,M,4] point form
    const float* kp_gt    = (const float*)d_in[5];   // [B,M,10]
    // d_in[6] = labels_gt: all ones (class 1), folded into the math.

    char* ws = (char*)d_ws;
    size_t off = 0;
    auto take = [&](size_t bytes) { char* p = ws + off; off = (off + bytes + 255) & ~(size_t)255; return p; };

    float*              lossc  = (float*)take((size_t)Bn * Pn * 4);       // 8 MB
    unsigned char*      match  = (unsigned char*)take((size_t)Bn * Pn);   // 2 MB
    char*               z0     = ws + off;                                 // memset region start
    unsigned long long* keys   = (unsigned long long*)take(Bn * Mn * 8);
    int*                numpos = (int*)take(Bn * 4);
    unsigned*           h1     = (unsigned*)take((size_t)Bn * 4096 * 4);
    unsigned*           h2     = (unsigned*)take((size_t)Bn * 4096 * 4);
    unsigned*           h3     = (unsigned*)take((size_t)Bn * 256 * 4);
    char*               z1     = ws + off;                                 // memset region end
    float*              p_ll   = (float*)take(Bn * 32 * 4);
    float*              p_lm   = (float*)take(Bn * 32 * 4);
    float*              p_ce   = (float*)take(Bn * 32 * 4);
    float*              p_neg  = (float*)take(Bn * 32 * 4);
    int*                t1a    = (int*)take(Bn * 4);
    int*                k1a    = (int*)take(Bn * 4);
    int*                t12a   = (int*)take(Bn * 4);
    int*                k2a    = (int*)take(Bn * 4);
    unsigned*           Tfin   = (unsigned*)take(Bn * 4);
    int*                eq     = (int*)take(Bn * 4);

    hipMemsetAsync(z0, 0, (size_t)(z1 - z0), stream);    // keys/numpos/histograms

    k_match   <<<dim3(Pn / 256, Bn), 256, 0, stream>>>(priors, boxes_gt, match, keys);
    k_override<<<1, 32, 0, stream>>>(keys, match);
    k_losses  <<<dim3(32, Bn), 256, 0, stream>>>(loc, conf, landm, priors, boxes_gt,
                                                 kp_gt, match, lossc, h1, p_ll, p_lm, p_ce, numpos);
    k_scan1   <<<1, 32, 0, stream>>>(h1, numpos, t1a, k1a, Tfin, eq);
    k_hist2   <<<dim3(32, Bn), 256, 0, stream>>>(lossc, t1a, h2);
    k_scan2   <<<1, 32, 0, stream>>>(h2, t1a, k1a, t12a, k2a);
    k_hist3   <<<dim3(32, Bn), 256, 0, stream>>>(lossc, t12a, h3);
    k_scan3   <<<1, 32, 0, stream>>>(h3, t12a, k2a, Tfin, eq);
    k_negsum  <<<dim3(32, Bn), 256, 0, stream>>>(lossc, Tfin, p_neg);
    k_combine <<<1, 32, 0, stream>>>(p_ll, p_lm, p_ce, p_neg, numpos, Tfin, eq, (float*)d_out);
}